// SelectiveSSM_54271206752703
// MI455X (gfx1250) — compile-verified
//
#include <hip/hip_runtime.h>
#include <hip/hip_bf16.h>

#define D_MODEL 1024
#define D_STATE 16
#define DT_RANK 64
#define BATCH   2
#define SEQLEN  4096
#define E_DIM   (DT_RANK + 2*D_STATE)   // 96
#define ROWS    (BATCH*SEQLEN)          // 8192
#define NCHUNK  64
#define TCHUNK  (SEQLEN/NCHUNK)         // 64

typedef __attribute__((ext_vector_type(2))) float v2f;
typedef __attribute__((ext_vector_type(8))) float v8f;

__device__ __forceinline__ float softplus_f(float v) {
  return v > 20.f ? v : log1pf(__expf(v));
}

// ---------------------------------------------------------------------------
// GEMM 1: dtBC[row, e] = sum_d x[row, d] * Wbc[e, d]
// One wave computes one 16x16 output tile via V_WMMA_F32_16X16X4_F32,
// K-loop over D_MODEL in steps of 4.
// A (16x4 f32): lanes 0-15 row M=l hold K=0,1; lanes 16-31 hold K=2,3.
// B (4x16 f32): lanes 0-15 col N=l hold K=0,1; lanes 16-31 hold K=2,3.
// D (16x16 f32): VGPR r = row (8*half + r), col = l.
// ---------------------------------------------------------------------------
__global__ __launch_bounds__(256) void k_proj_dtbc(const float* __restrict__ x,
                                                   const float* __restrict__ Wbc,
                                                   float* __restrict__ dtBC) {
  const int NT = E_DIM / 16; // 6
  int wave = (blockIdx.x * blockDim.x + threadIdx.x) >> 5;
  int lane = threadIdx.x & 31;
  int mt = wave / NT, nt = wave - mt * NT;
  int l = lane & 15, half = lane >> 4;

  const float* arow = x   + (size_t)(mt * 16 + l) * D_MODEL + 2 * half;
  const float* brow = Wbc + (size_t)(nt * 16 + l) * D_MODEL + 2 * half;

  v8f acc = {0.f, 0.f, 0.f, 0.f, 0.f, 0.f, 0.f, 0.f};
  for (int k = 0; k < D_MODEL; k += 4) {
    v2f a, b;
    a[0] = arow[k]; a[1] = arow[k + 1];
    b[0] = brow[k]; b[1] = brow[k + 1];
    acc = __builtin_amdgcn_wmma_f32_16x16x4_f32(false, a, false, b,
                                                (short)0, acc, false, false);
  }
#pragma unroll
  for (int r = 0; r < 8; ++r) {
    int m = mt * 16 + 8 * half + r;
    dtBC[(size_t)m * E_DIM + nt * 16 + l] = acc[r];
  }
}

// ---------------------------------------------------------------------------
// GEMM 2: dt[row, d] = softplus( sum_r dtBC[row, r] * Wdt[d, r] + bias[d] )
// M = 8192, N = 1024, K = 64 (first 64 columns of dtBC).
// ---------------------------------------------------------------------------
__global__ __launch_bounds__(256) void k_proj_dt(const float* __restrict__ dtBC,
                                                 const float* __restrict__ Wdt,
                                                 const float* __restrict__ bdt,
                                                 float* __restrict__ dt) {
  const int NT = D_MODEL / 16; // 64
  int wave = (blockIdx.x * blockDim.x + threadIdx.x) >> 5;
  int lane = threadIdx.x & 31;
  int mt = wave / NT, nt = wave - mt * NT;
  int l = lane & 15, half = lane >> 4;

  const float* arow = dtBC + (size_t)(mt * 16 + l) * E_DIM   + 2 * half;
  const float* brow = Wdt  + (size_t)(nt * 16 + l) * DT_RANK + 2 * half;

  v8f acc = {0.f, 0.f, 0.f, 0.f, 0.f, 0.f, 0.f, 0.f};
  for (int k = 0; k < DT_RANK; k += 4) {
    v2f a, b;
    a[0] = arow[k]; a[1] = arow[k + 1];
    b[0] = brow[k]; b[1] = brow[k + 1];
    acc = __builtin_amdgcn_wmma_f32_16x16x4_f32(false, a, false, b,
                                                (short)0, acc, false, false);
  }
  float bias = bdt[nt * 16 + l];
#pragma unroll
  for (int r = 0; r < 8; ++r) {
    int m = mt * 16 + 8 * half + r;
    dt[(size_t)m * D_MODEL + nt * 16 + l] = softplus_f(acc[r] + bias);
  }
}

// ---------------------------------------------------------------------------
// Scan pass 1 (chunk-local): one thread per (b, d, chunk).
// h_local starts at 0; also accumulates P[n] = prod_t exp(dt*A[n]).
// ---------------------------------------------------------------------------
__global__ __launch_bounds__(256) void k_scan_local(const float* __restrict__ x,
                                                    const float* __restrict__ dt,
                                                    const float* __restrict__ dtBC,
                                                    const float* __restrict__ A_log,
                                                    float* __restrict__ hend,
                                                    float* __restrict__ Pprod) {
  int gid = blockIdx.x * blockDim.x + threadIdx.x;
  int d = gid & (D_MODEL - 1);
  int c = (gid >> 10) & (NCHUNK - 1);
  int b = gid >> 16;

  float An[D_STATE];
#pragma unroll
  for (int n = 0; n < D_STATE; ++n) An[n] = -__expf(A_log[d * D_STATE + n]);

  float h[D_STATE], P[D_STATE];
#pragma unroll
  for (int n = 0; n < D_STATE; ++n) { h[n] = 0.f; P[n] = 1.f; }

  int t0 = c * TCHUNK;
  for (int t = t0; t < t0 + TCHUNK; ++t) {
    size_t row = (size_t)b * SEQLEN + t;
    float dtv = dt[row * D_MODEL + d];
    float xv  = x[row * D_MODEL + d];
    float Bv[D_STATE];
    const float4* bp = (const float4*)(dtBC + row * E_DIM + DT_RANK);
#pragma unroll
    for (int i = 0; i < 4; ++i) {
      float4 q = bp[i];
      Bv[4*i+0] = q.x; Bv[4*i+1] = q.y; Bv[4*i+2] = q.z; Bv[4*i+3] = q.w;
    }
    float dx = dtv * xv;
#pragma unroll
    for (int n = 0; n < D_STATE; ++n) {
      float dA = __expf(dtv * An[n]);
      h[n] = dA * h[n] + dx * Bv[n];
      P[n] *= dA;
    }
  }
  size_t base = ((size_t)(b * D_MODEL + d) * NCHUNK + c) * D_STATE;
#pragma unroll
  for (int n = 0; n < D_STATE; ++n) {
    hend[base + n]  = h[n];
    Pprod[base + n] = P[n];
  }
}

// ---------------------------------------------------------------------------
// Combine: one thread per (b, d, n); sequentially chain the 64 chunk states,
// writing each chunk's true initial state h0.
// ---------------------------------------------------------------------------
__global__ __launch_bounds__(256) void k_scan_combine(const float* __restrict__ hend,
                                                      const float* __restrict__ Pprod,
                                                      float* __restrict__ h0) {
  int gid = blockIdx.x * blockDim.x + threadIdx.x; // BATCH*D_MODEL*D_STATE
  int n = gid & (D_STATE - 1);
  int bd = gid >> 4;
  size_t base = (size_t)bd * NCHUNK * D_STATE + n;
  float h = 0.f;
  for (int c = 0; c < NCHUNK; ++c) {
    size_t idx = base + (size_t)c * D_STATE;
    h0[idx] = h;
    h = Pprod[idx] * h + hend[idx];
  }
}

// ---------------------------------------------------------------------------
// Scan pass 2: replay each chunk from its true h0, contract with C, emit y.
// ---------------------------------------------------------------------------
__global__ __launch_bounds__(256) void k_scan_emit(const float* __restrict__ x,
                                                   const float* __restrict__ dt,
                                                   const float* __restrict__ dtBC,
                                                   const float* __restrict__ A_log,
                                                   const float* __restrict__ Dp,
                                                   const float* __restrict__ h0,
                                                   float* __restrict__ y) {
  int gid = blockIdx.x * blockDim.x + threadIdx.x;
  int d = gid & (D_MODEL - 1);
  int c = (gid >> 10) & (NCHUNK - 1);
  int b = gid >> 16;

  float An[D_STATE];
#pragma unroll
  for (int n = 0; n < D_STATE; ++n) An[n] = -__expf(A_log[d * D_STATE + n]);

  float h[D_STATE];
  size_t base = ((size_t)(b * D_MODEL + d) * NCHUNK + c) * D_STATE;
#pragma unroll
  for (int n = 0; n < D_STATE; ++n) h[n] = h0[base + n];

  float Dd = Dp[d];
  int t0 = c * TCHUNK;
  for (int t = t0; t < t0 + TCHUNK; ++t) {
    size_t row = (size_t)b * SEQLEN + t;
    float dtv = dt[row * D_MODEL + d];
    float xv  = x[row * D_MODEL + d];
    float Bv[D_STATE], Cv[D_STATE];
    const float4* bp = (const float4*)(dtBC + row * E_DIM + DT_RANK);
    const float4* cp = (const float4*)(dtBC + row * E_DIM + DT_RANK + D_STATE);
#pragma unroll
    for (int i = 0; i < 4; ++i) {
      float4 q = bp[i];
      Bv[4*i+0] = q.x; Bv[4*i+1] = q.y; Bv[4*i+2] = q.z; Bv[4*i+3] = q.w;
    }
#pragma unroll
    for (int i = 0; i < 4; ++i) {
      float4 q = cp[i];
      Cv[4*i+0] = q.x; Cv[4*i+1] = q.y; Cv[4*i+2] = q.z; Cv[4*i+3] = q.w;
    }
    float dx = dtv * xv;
    float yv = 0.f;
#pragma unroll
    for (int n = 0; n < D_STATE; ++n) {
      float dA = __expf(dtv * An[n]);
      h[n] = dA * h[n] + dx * Bv[n];
      yv += h[n] * Cv[n];
    }
    y[row * D_MODEL + d] = yv + xv * Dd;
  }
}

extern "C" void kernel_launch(void* const* d_in, const int* in_sizes, int n_in,
                              void* d_out, int out_size, void* d_ws, size_t ws_size,
                              hipStream_t stream) {
  const float* x    = (const float*)d_in[0]; // (2, 4096, 1024)
  const float* Alog = (const float*)d_in[1]; // (1024, 16)
  const float* Dp   = (const float*)d_in[2]; // (1024,)
  const float* Wdt  = (const float*)d_in[3]; // (1024, 64)
  const float* bdt  = (const float*)d_in[4]; // (1024,)
  const float* Wbc  = (const float*)d_in[5]; // (96, 1024)
  float* y = (float*)d_out;                  // (2, 4096, 1024)

  float* ws    = (float*)d_ws;
  float* dtBC  = ws;                                   // 8192*96
  float* dt    = dtBC + (size_t)ROWS * E_DIM;          // 8192*1024
  float* hend  = dt + (size_t)ROWS * D_MODEL;          // 2*1024*64*16
  float* Pprod = hend + (size_t)BATCH * D_MODEL * NCHUNK * D_STATE;
  float* h0    = Pprod + (size_t)BATCH * D_MODEL * NCHUNK * D_STATE;

  // GEMM 1: 512 x 6 tiles -> 3072 waves -> 384 blocks of 8 waves
  int waves1 = (ROWS / 16) * (E_DIM / 16);
  k_proj_dtbc<<<waves1 * 32 / 256, 256, 0, stream>>>(x, Wbc, dtBC);

  // GEMM 2: 512 x 64 tiles -> 32768 waves -> 4096 blocks
  int waves2 = (ROWS / 16) * (D_MODEL / 16);
  k_proj_dt<<<waves2 * 32 / 256, 256, 0, stream>>>(dtBC, Wdt, bdt, dt);

  // Chunked scan
  int nscan = BATCH * D_MODEL * NCHUNK;   // 131072 threads
  k_scan_local<<<nscan / 256, 256, 0, stream>>>(x, dt, dtBC, Alog, hend, Pprod);

  int ncomb = BATCH * D_MODEL * D_STATE;  // 32768 threads
  k_scan_combine<<<ncomb / 256, 256, 0, stream>>>(hend, Pprod, h0);

  k_scan_emit<<<nscan / 256, 256, 0, stream>>>(x, dt, dtBC, Alog, Dp, h0, y);
}